// ImaginariumModel_69260642615609
// MI455X (gfx1250) — compile-verified
//
#include <hip/hip_runtime.h>
#include <math.h>

typedef __attribute__((ext_vector_type(2))) float v2f;
typedef __attribute__((ext_vector_type(4))) float v4f;
typedef __attribute__((ext_vector_type(8))) float v8f;

#define BPB        16      // batch rows per workgroup
#define NCARDS     64
#define NWORDS     50
#define LEMB       128
#define LDENSE     128
#define THREADS    128     // 4 waves (wave32)
#define XSTRIDE    132     // LDS pad: bank-conflict-free column reads
#define EPS        1e-8f

__global__ __launch_bounds__(THREADS)
void imaginarium_fused(const float* __restrict__ x_img,
                       const float* __restrict__ x_txt,
                       const float* __restrict__ y,
                       const float* __restrict__ W,     // [LDENSE][LEMB] row-major
                       const float* __restrict__ bias,  // [LDENSE]
                       float* __restrict__ out) {       // [B][NWORDS]
    __shared__ float Xg[BPB][XSTRIDE];        // gathered image rows
    __shared__ float Ld[BPB][XSTRIDE];        // projected leader vectors
    __shared__ float Lg[BPB][NWORDS];         // logits -> exp values
    __shared__ float naS[BPB];
    __shared__ float invS[BPB];
    __shared__ int   idxS[BPB];

    const int t     = threadIdx.x;
    const int lane  = t & 31;
    const int wid   = t >> 5;
    const int bbase = blockIdx.x * BPB;

    // ---- Phase 1: argmax over cards (first-max tie-break == jnp.argmax) ----
    if (t < BPB) {
        const float* yr = y + (size_t)(bbase + t) * NCARDS;
        float best = yr[0]; int bi = 0;
        for (int c = 1; c < NCARDS; ++c) {
            float v = yr[c];
            if (v > best) { best = v; bi = c; }
        }
        idxS[t] = bi;
    }
    __syncthreads();

    // ---- Phase 2: gather chosen rows into LDS (coalesced) ----
    for (int i = t; i < BPB * LEMB; i += THREADS) {
        int r = i >> 7, d = i & 127;
        Xg[r][d] = x_img[((size_t)(bbase + r) * NCARDS + idxS[r]) * LEMB + d];
    }
    __syncthreads();

    // ---- Phase 3: leader = Xg(16x128) @ W^T(128x128) + b via WMMA f32 16x16x4 ----
    // A 16x4 layout (ISA 7.12.2): lane<16 -> M=lane, K=k0+{0,1}; lane>=16 -> K=k0+{2,3}
    // B 4x16:                      lane&15 -> N,      same K pattern
    // C/D 16x16: VGPR r -> M = r + (lane>=16 ? 8 : 0), N = lane&15
    {
        const int m  = lane & 15;
        const int kh = (lane >> 4) << 1;          // 0 or 2
        #pragma unroll
        for (int tt = 0; tt < 2; ++tt) {
            const int e0 = wid * 32 + tt * 16;    // output-column tile
            v8f c = {};
            const float* wb = W + (size_t)(e0 + m) * LEMB + kh;  // B[d][e]=W[e][d]
            for (int k0 = 0; k0 < LEMB; k0 += 4) {
                v2f a, b;
                a.x = Xg[m][k0 + kh];
                a.y = Xg[m][k0 + kh + 1];
                b.x = wb[k0];
                b.y = wb[k0 + 1];
                c = __builtin_amdgcn_wmma_f32_16x16x4_f32(
                        false, a, false, b, (short)0, c, false, false);
            }
            const int   mb = (lane >> 4) * 8;
            const float bv = bias[e0 + m];
            #pragma unroll
            for (int r = 0; r < 8; ++r) Ld[mb + r][e0 + m] = c[r] + bv;
        }
    }
    __syncthreads();

    // ---- Phase 3.5: leader norms ----
    if (t < BPB) {
        float s = 0.f;
        for (int e = 0; e < LDENSE; ++e) { float v = Ld[t][e]; s = fmaf(v, v, s); }
        naS[t] = sqrtf(s);
    }
    __syncthreads();

    // ---- Phase 4: stream x_txt once; fused dot + norm + cosine ----
    // one wave per (batch,word) pair: 32 lanes x float4 = full 512B row, coalesced
    {
        const float* txt = x_txt + (size_t)bbase * NWORDS * LEMB;
        for (int p = wid; p < BPB * NWORDS; p += 4) {
            const int b = p / NWORDS;
            const int w = p - b * NWORDS;
            const float* xr = txt + (size_t)p * LEMB + lane * 4;
            if (p + 4 < BPB * NWORDS)
                __builtin_prefetch(txt + (size_t)(p + 4) * LEMB + lane * 4, 0, 3);
            v4f xv = *(const v4f*)xr;
            v4f lv = *(const v4f*)&Ld[b][lane * 4];
            float dot = 0.f, ss = 0.f;
            #pragma unroll
            for (int j = 0; j < 4; ++j) {
                dot = fmaf(xv[j], lv[j], dot);
                ss  = fmaf(xv[j], xv[j], ss);
            }
            #pragma unroll
            for (int off = 16; off >= 1; off >>= 1) {   // wave32 reduction
                dot += __shfl_xor(dot, off, 32);
                ss  += __shfl_xor(ss,  off, 32);
            }
            if (lane == 0)
                Lg[b][w] = dot / fmaxf(naS[b] * sqrtf(ss), EPS);
        }
    }
    __syncthreads();

    // ---- Phase 5: softmax over the 50 words per row ----
    if (t < BPB) {
        float mx = Lg[t][0];
        for (int w = 1; w < NWORDS; ++w) mx = fmaxf(mx, Lg[t][w]);
        float s = 0.f;
        for (int w = 0; w < NWORDS; ++w) {
            float e = expf(Lg[t][w] - mx);
            Lg[t][w] = e;
            s += e;
        }
        invS[t] = 1.0f / s;
    }
    __syncthreads();

    // contiguous 800-float block per workgroup
    float* ob = out + (size_t)bbase * NWORDS;
    for (int i = t; i < BPB * NWORDS; i += THREADS) {
        int b = i / NWORDS;
        ob[i] = Lg[b][i - b * NWORDS] * invS[b];
    }
}

extern "C" void kernel_launch(void* const* d_in, const int* in_sizes, int n_in,
                              void* d_out, int out_size, void* d_ws, size_t ws_size,
                              hipStream_t stream) {
    const float* x_img = (const float*)d_in[0];  // [B, 64, 128]
    const float* x_txt = (const float*)d_in[1];  // [B, 50, 128]
    const float* y     = (const float*)d_in[2];  // [B, 64]
    const float* W     = (const float*)d_in[3];  // [128, 128]
    const float* bias  = (const float*)d_in[4];  // [128]
    float* out = (float*)d_out;                  // [B, 50]

    const int B = in_sizes[2] / NCARDS;          // 16384
    const int grid = B / BPB;                    // 1024 workgroups
    imaginarium_fused<<<grid, THREADS, 0, stream>>>(x_img, x_txt, y, W, bias, out);
}